// GCNLayer_33552284516385
// MI455X (gfx1250) — compile-verified
//
#include <hip/hip_runtime.h>

typedef __attribute__((ext_vector_type(2))) float v2f;
typedef __attribute__((ext_vector_type(8))) float v8f;

#define IN_FEATS 256
#define OUT_FEATS 128

// ---------------------------------------------------------------------------
// Zero the aggregation buffer (graph-capture-safe, no hipMemset).
// ---------------------------------------------------------------------------
__global__ void __launch_bounds__(256) gcn_zero(float* __restrict__ p, int n) {
  int i = blockIdx.x * 256 + threadIdx.x;
  if (i < n) p[i] = 0.0f;
}

// ---------------------------------------------------------------------------
// hw = h @ W via V_WMMA_F32_16X16X4_F32 (fp32 in/out, no precision loss).
// Block = 128 threads = 4 waves covering a 16-row x 128-col strip.
// Each wave register-blocks TWO 16-col tiles (c0, c1): one A fragment feeds
// two WMMAs, halving redundant A loads vs one-tile-per-wave.
//
// VGPR layouts per CDNA5 ISA 7.12.2 (32-bit operands, wave32):
//   A (16x4):  lanes 0-15 hold K={0,1}, lanes 16-31 hold K={2,3}; M = lane&15
//   B (4x16):  lanes 0-15 hold K={0,1}, lanes 16-31 hold K={2,3}; N = lane&15
//   C/D:       VGPR r -> row r + 8*(lane>=16), col = lane&15
// ---------------------------------------------------------------------------
__global__ void __launch_bounds__(128) gcn_gemm_wmma(const float* __restrict__ h,
                                                     const float* __restrict__ W,
                                                     float* __restrict__ hw,
                                                     int nNodes) {
  const int m0   = blockIdx.x * 16;
  const int wave = threadIdx.x >> 5;   // 0..3 -> 32-col slab
  const int lane = threadIdx.x & 31;
  const int n0   = wave * 32;
  const int row  = lane & 15;
  const int half = lane >> 4;          // 0: K pair {0,1}; 1: K pair {2,3}

  int m = m0 + row;
  if (m >= nNodes) m = nNodes - 1;     // clamp loads in (never-taken) tail tile

  const float* __restrict__ hrow  = h + (size_t)m * IN_FEATS;
  const float* __restrict__ wcol0 = W + n0 + row;        // tile 0 column
  const float* __restrict__ wcol1 = W + n0 + 16 + row;   // tile 1 column

  v8f c0 = {};
  v8f c1 = {};
#pragma unroll 4
  for (int k = 0; k < IN_FEATS; k += 4) {
    const int kk = k + 2 * half;
    const v2f a = *reinterpret_cast<const v2f*>(hrow + kk);  // global_load_b64
    v2f b0, b1;
    b0.x = wcol0[(size_t)(kk + 0) * OUT_FEATS];
    b0.y = wcol0[(size_t)(kk + 1) * OUT_FEATS];
    b1.x = wcol1[(size_t)(kk + 0) * OUT_FEATS];
    b1.y = wcol1[(size_t)(kk + 1) * OUT_FEATS];
    // (neg_a, A, neg_b, B, c_mod, C, reuse_a, reuse_b)
    c0 = __builtin_amdgcn_wmma_f32_16x16x4_f32(false, a, false, b0, (short)0,
                                               c0, false, false);
    c1 = __builtin_amdgcn_wmma_f32_16x16x4_f32(false, a, false, b1, (short)0,
                                               c1, false, false);
  }

  float* __restrict__ o = hw + (size_t)m0 * OUT_FEATS + row;
  if (m0 + 16 <= nNodes) {
    // Fast path (always taken when nNodes % 16 == 0): unpredicated stores.
#pragma unroll
    for (int r = 0; r < 8; ++r) {
      const int mr = r + 8 * half;
      o[(size_t)mr * OUT_FEATS + n0]      = c0[r];
      o[(size_t)mr * OUT_FEATS + n0 + 16] = c1[r];
    }
  } else {
#pragma unroll
    for (int r = 0; r < 8; ++r) {
      const int mr = r + 8 * half;
      if (m0 + mr < nNodes) {
        o[(size_t)mr * OUT_FEATS + n0]      = c0[r];
        o[(size_t)mr * OUT_FEATS + n0 + 16] = c1[r];
      }
    }
  }
}

// ---------------------------------------------------------------------------
// Edge scatter: one wave32 per edge. src/dst indices are forced wave-uniform
// via readfirstlane so the compiler emits scalar (s_load) index fetches; each
// lane then moves a float4 (512 B per wave, coalesced, L2-resident) and
// issues 4 no-return global_atomic_add_f32 into the destination row.
// This phase is the bandwidth roofline limiter of the whole layer.
// ---------------------------------------------------------------------------
__global__ void __launch_bounds__(256) gcn_scatter(const float* __restrict__ hw,
                                                   const int* __restrict__ src,
                                                   const int* __restrict__ dst,
                                                   float* __restrict__ agg,
                                                   int nEdges) {
  int i = blockIdx.x * 256 + threadIdx.x;
  int e = __builtin_amdgcn_readfirstlane(i >> 5);  // wave-uniform edge id
  if (e >= nEdges) return;
  int g = (i & 31) << 2;                           // lane's 4-float group

  const int s = src[e];                            // scalar loads (KMcnt)
  const int d = dst[e];

  const float4 v =
      *reinterpret_cast<const float4*>(hw + (size_t)s * OUT_FEATS + g);
  float* o = agg + (size_t)d * OUT_FEATS + g;
  atomicAdd(o + 0, v.x);
  atomicAdd(o + 1, v.y);
  atomicAdd(o + 2, v.z);
  atomicAdd(o + 3, v.w);
}

// ---------------------------------------------------------------------------
// out = relu(agg + b), in place.
// ---------------------------------------------------------------------------
__global__ void __launch_bounds__(256) gcn_epilogue(float* __restrict__ out,
                                                    const float* __restrict__ bias,
                                                    int n) {
  int i = blockIdx.x * 256 + threadIdx.x;
  if (i < n) {
    float v = out[i] + bias[i & (OUT_FEATS - 1)];
    out[i] = v > 0.0f ? v : 0.0f;
  }
}

extern "C" void kernel_launch(void* const* d_in, const int* in_sizes, int n_in,
                              void* d_out, int out_size, void* d_ws, size_t ws_size,
                              hipStream_t stream) {
  const float* h   = (const float*)d_in[0];  // [N, 256]
  const float* W   = (const float*)d_in[1];  // [256, 128]
  const float* b   = (const float*)d_in[2];  // [128]
  const int*   src = (const int*)d_in[3];    // [E]
  const int*   dst = (const int*)d_in[4];    // [E]
  float* out = (float*)d_out;                // [N, 128] (doubles as agg buffer)
  float* hw  = (float*)d_ws;                 // [N, 128] scratch: h @ W

  const int nNodes   = in_sizes[0] / IN_FEATS;
  const int nEdges   = in_sizes[3];
  const int outElems = nNodes * OUT_FEATS;

  // 1) agg = 0
  gcn_zero<<<(outElems + 255) / 256, 256, 0, stream>>>(out, outElems);

  // 2) hw = h @ W   (WMMA f32, 2 tiles per wave)
  gcn_gemm_wmma<<<(nNodes + 15) / 16, 128, 0, stream>>>(h, W, hw, nNodes);

  // 3) agg[dst] += hw[src]   (one wave per edge)
  const long long scatterThreads = (long long)nEdges * 32;
  gcn_scatter<<<(int)((scatterThreads + 255) / 256), 256, 0, stream>>>(
      hw, src, dst, out, nEdges);

  // 4) out = relu(agg + b)
  gcn_epilogue<<<(outElems + 255) / 256, 256, 0, stream>>>(out, b, outElems);
}